// S6_89438398972449
// MI455X (gfx1250) — compile-verified
//
#include <hip/hip_runtime.h>
#include <math.h>

typedef float v2f __attribute__((ext_vector_type(2)));
typedef float v8f __attribute__((ext_vector_type(8)));

#define S6_DIM    1024
#define S6_HID    2048
#define S6_DST    16
#define S6_DRANK  64
#define S6_KCONV  4
#define S6_N      2
#define S6_T      1024
#define S6_M      (S6_N * S6_T)   // 2048 rows total

#if __has_builtin(__builtin_amdgcn_global_load_async_to_lds_b128) && \
    __has_builtin(__builtin_amdgcn_s_wait_asynccnt)
#define S6_HAS_ASYNC 1
#else
#define S6_HAS_ASYNC 0
#endif

#if S6_HAS_ASYNC
// The builtin's param 0 is 'int __vector(4) __device__ *' (global int4 ptr),
// param 1 the LDS-side analog. Match with vector_size(16) int typedefs.
typedef int s6_v4i __attribute__((vector_size(16)));
typedef __attribute__((address_space(1))) s6_v4i* s6_gv4p;   // global int4*
typedef __attribute__((address_space(3))) s6_v4i* s6_lv4p;   // LDS    int4*
#endif

// ---------------------------------------------------------------------------
// fp32 WMMA helpers (V_WMMA_F32_16X16X4_F32, ISA 7.12.2 layouts)
//   A (16x4): lane m=lane&15; lanes 0-15 hold K=kb,kb+1; lanes 16-31 kb+2,kb+3
//   B (4x16): lane n=lane&15; same K split across lane halves
//   C/D: VGPR r -> M=r (lanes 0-15) / r+8 (lanes 16-31), N=lane&15
// ---------------------------------------------------------------------------
__device__ __forceinline__ v8f s6_wmma(v2f a, v2f b, v8f c) {
    return __builtin_amdgcn_wmma_f32_16x16x4_f32(false, a, false, b,
                                                 (short)0, c, false, false);
}

struct acc4 { v8f c00, c01, c10, c11; };   // [Mhalf][Nhalf] 16x16 subtiles

// 32x32 register-blocked tile: out[row0:+32, col0:+32] = A * W^T
// A: MxK row-major (lda), W: NxK row-major (ldw)  => B[k][n] = W[n][k]
// 4 fragment loads -> 4 WMMAs per K-step (2x the WMMA/VMEM density of 16x16).
__device__ __forceinline__ acc4 s6_wmma_tile32(const float* __restrict__ A, int lda,
                                               const float* __restrict__ W, int ldw,
                                               int kdim, int row0, int col0) {
    const int lane = threadIdx.x & 31;
    const int half = lane >> 4;
    const int lm   = lane & 15;
    const float* __restrict__ ar0 = A + (size_t)(row0 + lm) * lda;
    const float* __restrict__ ar1 = ar0 + (size_t)16 * lda;
    const float* __restrict__ wr0 = W + (size_t)(col0 + lm) * ldw;
    const float* __restrict__ wr1 = wr0 + (size_t)16 * ldw;
    acc4 acc = {};
    for (int kb = 0; kb < kdim; kb += 4) {
        const int k = kb + 2 * half;
        v2f a0 = { ar0[k], ar0[k + 1] };
        v2f a1 = { ar1[k], ar1[k + 1] };
        v2f b0 = { wr0[k], wr0[k + 1] };
        v2f b1 = { wr1[k], wr1[k + 1] };
        acc.c00 = s6_wmma(a0, b0, acc.c00);
        acc.c01 = s6_wmma(a0, b1, acc.c01);
        acc.c10 = s6_wmma(a1, b0, acc.c10);
        acc.c11 = s6_wmma(a1, b1, acc.c11);
    }
    return acc;
}

__device__ __forceinline__ void s6_store16(float* __restrict__ out, int ldo,
                                           int row0, int col0, int half, int lm, v8f c) {
#pragma unroll
    for (int r = 0; r < 8; ++r)
        out[(size_t)(row0 + r + 8 * half) * ldo + col0 + lm] = c[r];
}

// ---------------------------------------------------------------------------
// GEMM 1: uo = x @ W_in^T ; columns 0..2047 -> u_pre, 2048..4095 -> o
// ---------------------------------------------------------------------------
__global__ void s6_gemm_in(const float* __restrict__ x, const float* __restrict__ W_in,
                           float* __restrict__ u_pre, float* __restrict__ o_buf) {
    const int Mt = S6_M / 32, Nt = (2 * S6_HID) / 32;       // 64 x 128
    const int tile = blockIdx.x * 8 + (threadIdx.x >> 5);
    if (tile >= Mt * Nt) return;
    const int row0 = (tile % Mt) * 32, col0 = (tile / Mt) * 32;
    acc4 a = s6_wmma_tile32(x, S6_DIM, W_in, S6_DIM, S6_DIM, row0, col0);
    const int lane = threadIdx.x & 31, half = lane >> 4, lm = lane & 15;
    v8f cs[2][2] = { { a.c00, a.c01 }, { a.c10, a.c11 } };
#pragma unroll
    for (int i = 0; i < 2; ++i)
#pragma unroll
        for (int j = 0; j < 2; ++j) {
            const int rs = row0 + 16 * i, cn = col0 + 16 * j;
            if (cn < S6_HID) s6_store16(u_pre, S6_HID, rs, cn, half, lm, cs[i][j]);
            else             s6_store16(o_buf, S6_HID, rs, cn - S6_HID, half, lm, cs[i][j]);
        }
}

// ---------------------------------------------------------------------------
// Depthwise causal conv (K=4) + bias + SiLU
// ---------------------------------------------------------------------------
__global__ void s6_conv_silu(const float* __restrict__ u_pre,
                             const float* __restrict__ conv_w,
                             const float* __restrict__ conv_b,
                             float* __restrict__ u) {
    const int idx = blockIdx.x * 256 + threadIdx.x;         // over M*HID
    if (idx >= S6_M * S6_HID) return;
    const int h = idx & (S6_HID - 1);
    const int m = idx >> 11;                                // n*T + t
    const int t = m & (S6_T - 1);
    float s = conv_b[h];
#pragma unroll
    for (int j = 0; j < S6_KCONV; ++j) {
        const int ti = t - (S6_KCONV - 1) + j;
        if (ti >= 0)
            s += u_pre[(size_t)(m - (S6_KCONV - 1) + j) * S6_HID + h] * conv_w[h * S6_KCONV + j];
    }
    u[idx] = s * (1.0f / (1.0f + __expf(-s)));              // SiLU
}

// ---------------------------------------------------------------------------
// GEMM 2: g = u @ W_gate^T (N=96) -> delta_lo(0..63) / b(64..79) / c(80..95)
// 16-col subtile boundaries align with the 64/80 splits.
// ---------------------------------------------------------------------------
__global__ void s6_gemm_gate(const float* __restrict__ u, const float* __restrict__ W_gate,
                             float* __restrict__ dlo, float* __restrict__ bb,
                             float* __restrict__ cc) {
    const int Mt = S6_M / 32, Nt = (S6_DRANK + 2 * S6_DST) / 32;  // 64 x 3
    const int tile = blockIdx.x * 8 + (threadIdx.x >> 5);
    if (tile >= Mt * Nt) return;
    const int row0 = (tile % Mt) * 32, col0 = (tile / Mt) * 32;
    acc4 a = s6_wmma_tile32(u, S6_HID, W_gate, S6_HID, S6_HID, row0, col0);
    const int lane = threadIdx.x & 31, half = lane >> 4, lm = lane & 15;
    v8f cs[2][2] = { { a.c00, a.c01 }, { a.c10, a.c11 } };
#pragma unroll
    for (int i = 0; i < 2; ++i)
#pragma unroll
        for (int j = 0; j < 2; ++j) {
            const int rs = row0 + 16 * i, cn = col0 + 16 * j;
            if      (cn < S6_DRANK)          s6_store16(dlo, S6_DRANK, rs, cn, half, lm, cs[i][j]);
            else if (cn < S6_DRANK + S6_DST) s6_store16(bb, S6_DST, rs, cn - S6_DRANK, half, lm, cs[i][j]);
            else                             s6_store16(cc, S6_DST, rs, cn - S6_DRANK - S6_DST, half, lm, cs[i][j]);
        }
}

// ---------------------------------------------------------------------------
// GEMM 3: delta = softplus(dlo @ W_delta^T + delta_bias)
// ---------------------------------------------------------------------------
__global__ void s6_gemm_delta(const float* __restrict__ dlo, const float* __restrict__ W_delta,
                              const float* __restrict__ dbias, float* __restrict__ delta) {
    const int Mt = S6_M / 32, Nt = S6_HID / 32;             // 64 x 64
    const int tile = blockIdx.x * 8 + (threadIdx.x >> 5);
    if (tile >= Mt * Nt) return;
    const int row0 = (tile % Mt) * 32, col0 = (tile / Mt) * 32;
    acc4 a = s6_wmma_tile32(dlo, S6_DRANK, W_delta, S6_DRANK, S6_DRANK, row0, col0);
    const int lane = threadIdx.x & 31, half = lane >> 4, lm = lane & 15;
    v8f cs[2][2] = { { a.c00, a.c01 }, { a.c10, a.c11 } };
#pragma unroll
    for (int i = 0; i < 2; ++i)
#pragma unroll
        for (int j = 0; j < 2; ++j) {
            const int rs = row0 + 16 * i;
            const int cn = col0 + 16 * j + lm;
            const float bias = dbias[cn];
#pragma unroll
            for (int r = 0; r < 8; ++r) {
                float v = cs[i][j][r] + bias;
                v = (v > 20.0f) ? v : log1pf(__expf(v));    // softplus
                delta[(size_t)(rs + r + 8 * half) * S6_HID + cn] = v;
            }
        }
}

// ---------------------------------------------------------------------------
// Selective scan: one lane per (n,h); 16 states in registers.
// b/c (16 floats per (n,t), shared by all 2048 channels of a batch) staged
// through LDS in 64-step chunks, double-buffered; prefetched with
// GLOBAL_LOAD_ASYNC_TO_LDS_B128 under ASYNCcnt so the next chunk's copy
// overlaps the exp-heavy recurrence of the current chunk.
// ---------------------------------------------------------------------------
__global__ void s6_scan(const float* __restrict__ delta, const float* __restrict__ u,
                        const float* __restrict__ bb, const float* __restrict__ cc,
                        const float* __restrict__ A_log, float* __restrict__ y) {
    __shared__ float sb[2][64 * S6_DST];
    __shared__ float sc[2][64 * S6_DST];
    const int tid = threadIdx.x;
    const int gid = blockIdx.x * 256 + tid;                 // 0..4095
    const int n = gid >> 11;                                // block-uniform
    const int h = gid & (S6_HID - 1);
    float Ar[S6_DST], hs[S6_DST];
#pragma unroll
    for (int s = 0; s < S6_DST; ++s) {
        Ar[s] = -__expf(A_log[h * S6_DST + s]);
        hs[s] = 0.0f;
    }

#if S6_HAS_ASYNC
    // each thread copies 16B of b and 16B of c: 256 thr * 4 fl = one 1024-fl chunk
#define S6_ISSUE(buf, tb)                                                          \
    do {                                                                           \
        const size_t _base = (size_t)(n * S6_T + (tb)) * S6_DST + tid * 4;         \
        __builtin_amdgcn_global_load_async_to_lds_b128(                            \
            (s6_gv4p)(bb + _base), (s6_lv4p)&sb[(buf)][tid * 4], 0, 0);            \
        __builtin_amdgcn_global_load_async_to_lds_b128(                            \
            (s6_gv4p)(cc + _base), (s6_lv4p)&sc[(buf)][tid * 4], 0, 0);            \
    } while (0)
    S6_ISSUE(0, 0);
#else
    {
        const size_t base = (size_t)(n * S6_T) * S6_DST;
        for (int i = tid; i < 64 * S6_DST; i += 256) {
            sb[0][i] = bb[base + i];
            sc[0][i] = cc[base + i];
        }
    }
#endif

    const int NCH = S6_T / 64;                              // 16 chunks
    for (int cb = 0; cb < NCH; ++cb) {
        const int buf = cb & 1;
#if S6_HAS_ASYNC
        if (cb + 1 < NCH) {
            S6_ISSUE(buf ^ 1, (cb + 1) * 64);               // prefetch next chunk
            __builtin_amdgcn_s_wait_asynccnt(2);            // current chunk landed
        } else {
            __builtin_amdgcn_s_wait_asynccnt(0);
        }
#endif
        __syncthreads();                                    // staged data visible
        for (int tt = 0; tt < 64; ++tt) {
            const size_t m = (size_t)(n * S6_T + cb * 64 + tt);
            const float d  = delta[m * S6_HID + h];
            const float du = d * u[m * S6_HID + h];
            float acc = 0.0f;
#pragma unroll
            for (int s = 0; s < S6_DST; ++s) {
                const float f  = __expf(Ar[s] * d);
                const float hv = f * hs[s] + du * sb[buf][tt * S6_DST + s];
                hs[s] = hv;
                acc += hv * sc[buf][tt * S6_DST + s];
            }
            y[m * S6_HID + h] = acc;
        }
        __syncthreads();                                    // chunk consumed
#if !S6_HAS_ASYNC
        if (cb + 1 < NCH) {
            const size_t base = (size_t)(n * S6_T + (cb + 1) * 64) * S6_DST;
            for (int i = tid; i < 64 * S6_DST; i += 256) {
                sb[buf ^ 1][i] = bb[base + i];
                sc[buf ^ 1][i] = cc[base + i];
            }
        }
#endif
    }
}

// ---------------------------------------------------------------------------
// Gating: yg = (y + D*u) * sigmoid(o) * o
// ---------------------------------------------------------------------------
__global__ void s6_gating(const float* __restrict__ y, const float* __restrict__ u,
                          const float* __restrict__ o, const float* __restrict__ Dv,
                          float* __restrict__ yg) {
    const int idx = blockIdx.x * 256 + threadIdx.x;
    if (idx >= S6_M * S6_HID) return;
    const int h = idx & (S6_HID - 1);
    const float ov = o[idx];
    const float g = ov / (1.0f + __expf(-ov));              // sigmoid(o)*o
    yg[idx] = (y[idx] + Dv[h] * u[idx]) * g;
}

// ---------------------------------------------------------------------------
// GEMM 4: out = yg @ W_out^T
// ---------------------------------------------------------------------------
__global__ void s6_gemm_out(const float* __restrict__ yg, const float* __restrict__ W_out,
                            float* __restrict__ out) {
    const int Mt = S6_M / 32, Nt = S6_DIM / 32;             // 64 x 32
    const int tile = blockIdx.x * 8 + (threadIdx.x >> 5);
    if (tile >= Mt * Nt) return;
    const int row0 = (tile % Mt) * 32, col0 = (tile / Mt) * 32;
    acc4 a = s6_wmma_tile32(yg, S6_HID, W_out, S6_HID, S6_HID, row0, col0);
    const int lane = threadIdx.x & 31, half = lane >> 4, lm = lane & 15;
    v8f cs[2][2] = { { a.c00, a.c01 }, { a.c10, a.c11 } };
#pragma unroll
    for (int i = 0; i < 2; ++i)
#pragma unroll
        for (int j = 0; j < 2; ++j)
            s6_store16(out, S6_DIM, row0 + 16 * i, col0 + 16 * j, half, lm, cs[i][j]);
}

// ---------------------------------------------------------------------------
extern "C" void kernel_launch(void* const* d_in, const int* in_sizes, int n_in,
                              void* d_out, int out_size, void* d_ws, size_t ws_size,
                              hipStream_t stream) {
    const float* x      = (const float*)d_in[0];
    const float* W_in   = (const float*)d_in[1];
    const float* conv_w = (const float*)d_in[2];
    const float* conv_b = (const float*)d_in[3];
    const float* W_gate = (const float*)d_in[4];
    const float* W_delta= (const float*)d_in[5];
    const float* dbias  = (const float*)d_in[6];
    const float* A_log  = (const float*)d_in[7];
    const float* Dv     = (const float*)d_in[8];
    const float* W_out  = (const float*)d_in[9];
    float* out = (float*)d_out;

    float* ws = (float*)d_ws;
    const size_t BIG = (size_t)S6_M * S6_HID;               // 4,194,304 floats
    float* slot0 = ws;                                      // u_pre -> delta -> yg
    float* o_buf = ws + BIG;
    float* u_buf = ws + 2 * BIG;
    float* y_buf = ws + 3 * BIG;
    float* b_buf = ws + 4 * BIG;
    float* c_buf = b_buf + (size_t)S6_M * S6_DST;
    float* dlo   = c_buf + (size_t)S6_M * S6_DST;

    const int EW_BLOCKS = (S6_M * S6_HID + 255) / 256;      // 16384

    // 1) uo = x @ W_in^T        (32x32 tiles: 64*128 = 8192 waves -> 1024 blocks)
    s6_gemm_in<<<1024, 256, 0, stream>>>(x, W_in, slot0, o_buf);
    // 2) depthwise conv + SiLU
    s6_conv_silu<<<EW_BLOCKS, 256, 0, stream>>>(slot0, conv_w, conv_b, u_buf);
    // 3) g = u @ W_gate^T       (64*3 = 192 waves -> 24 blocks)
    s6_gemm_gate<<<24, 256, 0, stream>>>(u_buf, W_gate, dlo, b_buf, c_buf);
    // 4) delta = softplus(...)  (64*64 = 4096 waves -> 512 blocks)
    s6_gemm_delta<<<512, 256, 0, stream>>>(dlo, W_delta, dbias, slot0);
    // 5) sequential scan        (4096 channels -> 16 blocks)
    s6_scan<<<16, 256, 0, stream>>>(slot0, u_buf, b_buf, c_buf, A_log, y_buf);
    // 6) gating
    s6_gating<<<EW_BLOCKS, 256, 0, stream>>>(y_buf, u_buf, o_buf, Dv, slot0);
    // 7) out = yg @ W_out^T     (64*32 = 2048 waves -> 256 blocks)
    s6_gemm_out<<<256, 256, 0, stream>>>(slot0, W_out, out);
}